// RotatedGaussianAssigner_26130581029062
// MI455X (gfx1250) — compile-verified
//
#include <hip/hip_runtime.h>
#include <hip/hip_bf16.h>

typedef float v2f __attribute__((ext_vector_type(2)));
typedef float v8f __attribute__((ext_vector_type(8)));

namespace {
constexpr int   kC        = 15;      // classes
constexpr int   kNgt      = 64;      // GT boxes per batch
constexpr int   kL        = 21824;   // anchors total
constexpr float kEps      = 1e-9f;
constexpr float kGauss    = 12.0f;
constexpr float kScoreTh  = 0.23f;
constexpr int   kPS       = 20;      // LDS floats per GT (18 used, padded to 80B)
constexpr int   kWaves    = 4;       // waves per block (wave32)
constexpr int   kAnchWave = 16;      // anchors per wave tile
constexpr int   kAnchBlk  = kWaves * kAnchWave;  // 64
constexpr int   kBlkPerB  = kL / kAnchBlk;       // 341 (exact)
}

// ---- CDNA5 async global->LDS copy (ASYNCcnt-tracked, no VGPR data path) ----
__device__ __forceinline__ void async_copy_b32(void* lds_dst, const float* gaddr) {
  unsigned lds_off = (unsigned)(unsigned long long)lds_dst;  // addrspace(3) byte offset
  asm volatile("global_load_async_to_lds_b32 %0, %1, off"
               :: "v"(lds_off), "v"(gaddr)
               : "memory");
}
__device__ __forceinline__ void wait_asynccnt0() {
  asm volatile("s_wait_asynccnt 0x0" ::: "memory");
}

// lane k <-> lane k+16 exchange via LDS-permute hardware (SWAPX16)
__device__ __forceinline__ float swapx16f(float v) {
  return __builtin_bit_cast(float,
      __builtin_amdgcn_ds_swizzle(__builtin_bit_cast(int, v), 0x401f));
}
__device__ __forceinline__ int swapx16i(int v) {
  return __builtin_amdgcn_ds_swizzle(v, 0x401f);
}

// FCOS-style regress ranges by anchor level (cum counts 16384/20480/21504/21760/21824)
__device__ __forceinline__ void rangeOf(int l, float& lo, float& hi) {
  if (l < 16384)      { lo = -1.f;  hi = 64.f;  }
  else if (l < 20480) { lo = 64.f;  hi = 128.f; }
  else if (l < 21504) { lo = 128.f; hi = 256.f; }
  else if (l < 21760) { lo = 256.f; hi = 512.f; }
  else                { lo = 512.f; hi = 1e8f;  }
}

__device__ __forceinline__ void rbox2gauss(const float* rb, float& x, float& y,
                                           float& ga, float& gb, float& gc) {
  float ca = __cosf(rb[4]), sa = __sinf(rb[4]);
  float w2 = rb[2]*rb[2]*(1.f/12.f), h2 = rb[3]*rb[3]*(1.f/12.f);
  x = rb[0]; y = rb[1];
  ga = w2*ca*ca + h2*sa*sa;
  gb = w2*sa*sa + h2*ca*ca;
  gc = (w2 - h2)*ca*sa;
}

__device__ __forceinline__ float probiou(const float* b1, const float* b2) {
  constexpr float e = 1e-3f;
  float x1,y1,a1,bb1,c1, x2,y2,a2,bb2,c2;
  rbox2gauss(b1, x1,y1,a1,bb1,c1);
  rbox2gauss(b2, x2,y2,a2,bb2,c2);
  float dx = x1-x2, dy = y1-y2;
  float aa = a1+a2, bb = bb1+bb2, cc = c1+c2;
  float denom = aa*bb - cc*cc + e;
  float t1 = 0.25f*(aa*dy*dy + bb*dx*dx)/denom;
  float t2 = 0.5f*(cc*(-dx*dy))/denom;
  float det1 = fmaxf(a1*bb1 - c1*c1, 0.f);
  float det2 = fmaxf(a2*bb2 - c2*c2, 0.f);
  float t3 = 0.5f*__logf(denom/(4.f*__builtin_sqrtf(det1*det2) + e) + e);
  float bd = fminf(fmaxf(t1+t2+t3, e), 100.f);
  return 1.f - __builtin_sqrtf(1.f - __expf(-bd) + e);
}

__global__ __launch_bounds__(128)
void rot_gauss_assigner(const float* __restrict__ pred_boxes,   // (B,L,5)
                        const float* __restrict__ anchors,      // (1,L,2)
                        const int*   __restrict__ gt_labels,    // (B,64,1)
                        const float* __restrict__ gt_boxes,     // (B,64,5)
                        const float* __restrict__ pad_mask,     // (B,64,1)
                        const int*   __restrict__ bg_index_p,   // scalar
                        float* __restrict__ out, int Btot) {
  __shared__ float sgRaw[kNgt * 5];       // raw gt_boxes staged via async-LDS
  __shared__ float sg[kNgt * kPS];        // derived per-GT constants

  const int tid  = threadIdx.x;
  const int lane = tid & 31;
  const int wave = tid >> 5;
  const int b    = blockIdx.x / kBlkPerB;
  const int tg   = blockIdx.x % kBlkPerB;

  // ---------------- async-stage raw GT boxes into LDS (320 dwords) ----------
  {
    const float* gbase = gt_boxes + (size_t)b * kNgt * 5;
    async_copy_b32(&sgRaw[tid],       gbase + tid);
    async_copy_b32(&sgRaw[tid + 128], gbase + tid + 128);
    if (tid < kNgt) async_copy_b32(&sgRaw[tid + 256], gbase + tid + 256);
  }
  wait_asynccnt0();
  __syncthreads();

  // ---------------- derive per-GT constants into LDS ----------------
  if (tid < kNgt) {
    const int n = tid;
    const float* rb = sgRaw + n*5;
    float cx = rb[0], cy = rb[1], w = rb[2], h = rb[3], ang = rb[4];
    float ca = __cosf(ang), sa = __sinf(ang);
    // corner a of the poly; edge vectors ab (width dir) and ad (height dir)
    float ax = cx - 0.5f*w*ca + 0.5f*h*sa;
    float ay = cy - 0.5f*w*sa - 0.5f*h*ca;
    float vabx =  w*ca, vaby = w*sa;
    float vadx = -h*sa, vady = h*ca;
    float nab2 = vabx*vabx + vaby*vaby;
    float nad2 = vadx*vadx + vady*vady;
    float nab  = __builtin_sqrtf(nab2 + kEps);
    float nad  = __builtin_sqrtf(nad2 + kEps);
    float me   = fminf(w, h);
    float invA = 1.f / (nab*nab*nab*me + kEps);
    float invB = 1.f / (nad*nad*nad*me + kEps);
    float sigma = me * (1.f/kGauss);
    float invR = 1.f / (2.f*3.14159265358979323846f*sigma + kEps);
    float aca = __builtin_fabsf(ca), asa = __builtin_fabsf(sa);
    float ew = 0.5f*(w*aca + h*asa);
    float eh = 0.5f*(w*asa + h*aca);
    float* g = sg + n*kPS;
    g[0]  = vabx; g[1] = vaby; g[2] = vadx; g[3] = vady;
    g[4]  = -(cx*vabx + cy*vaby);   // const term of ctp.vab
    g[5]  = -(cx*vadx + cy*vady);   // const term of ctp.vad
    g[6]  = -(ax*vabx + ay*vaby);   // const term of ap.vab
    g[7]  = -(ax*vadx + ay*vady);   // const term of ap.vad
    g[8]  = invA; g[9] = invB; g[10] = invR;
    g[11] = nab2; g[12] = nad2;
    g[13] = cx - ew; g[14] = cy - eh; g[15] = cx + ew; g[16] = cy + eh;
    g[17] = pad_mask[b*kNgt + n];
  }
  __syncthreads();

  // ---------------- per-wave 16-anchor tile ----------------
  const int col = lane & 15;                 // anchor column N (0..15)
  const int l0  = tg*kAnchBlk + wave*kAnchWave;
  const int l   = l0 + col;
  const float px = anchors[2*l];
  const float py = anchors[2*l + 1];
  float rlo, rhi; rangeOf(l, rlo, rhi);

  const bool lowHalf = lane < 16;
  // B matrix 4x16: K=0 row = px, K=1 row = py, K=2 row = 1, K=3 row = 0
  // (lanes 0-15 carry K=0/1 in VGPR0/1; lanes 16-31 carry K=2/3)
  v2f bmat;
  bmat.x = lowHalf ? px : 1.0f;
  bmat.y = lowHalf ? py : 0.0f;

  __builtin_prefetch(pred_boxes + (size_t)((size_t)b*kL + l)*5, 0, 0);

  float bestv = -3.0e38f;
  int   besti = 0;

  #pragma unroll
  for (int tile = 0; tile < 4; ++tile) {
    // A matrices 16x4: row M = [vx, vy, c, 0]. Load coeff row unconditionally
    // (two b128 LDS loads), select in registers -> no saveexec predication.
    const int gm = tile*16 + col;            // this lane's A row M
    const float* g = sg + gm*kPS;
    const float4 g03 = *(const float4*)(g);      // vabx vaby vadx vady
    const float4 g47 = *(const float4*)(g + 4);  // cCtpAb cCtpAd cApAb cApAd
    v2f aCtpAb, aCtpAd, aApAb, aApAd;
    aCtpAb.x = lowHalf ? g03.x : g47.x;  aCtpAb.y = lowHalf ? g03.y : 0.f;
    aCtpAd.x = lowHalf ? g03.z : g47.y;  aCtpAd.y = lowHalf ? g03.w : 0.f;
    aApAb.x  = lowHalf ? g03.x : g47.z;  aApAb.y  = lowHalf ? g03.y : 0.f;
    aApAd.x  = lowHalf ? g03.z : g47.w;  aApAd.y  = lowHalf ? g03.w : 0.f;

    v8f dAB = {}, dAD = {}, pAB = {}, pAD = {};
    dAB = __builtin_amdgcn_wmma_f32_16x16x4_f32(false, aCtpAb, false, bmat,
                                                (short)0, dAB, false, false);
    dAD = __builtin_amdgcn_wmma_f32_16x16x4_f32(false, aCtpAd, false, bmat,
                                                (short)0, dAD, false, false);
    pAB = __builtin_amdgcn_wmma_f32_16x16x4_f32(false, aApAb, false, bmat,
                                                (short)0, pAB, false, false);
    pAD = __builtin_amdgcn_wmma_f32_16x16x4_f32(false, aApAd, false, bmat,
                                                (short)0, pAD, false, false);

    // C/D layout: VGPR r holds (M = r [+8 for hi half], N = lane%16)
    const int half = lowHalf ? 0 : 8;
    #pragma unroll
    for (int r = 0; r < 8; ++r) {
      const int m = tile*16 + half + r;
      const float* q = sg + m*kPS;
      const float4 qa = *(const float4*)(q + 8);   // invA invB invR nab2
      const float4 qb = *(const float4*)(q + 12);  // nad2 x1 y1 x2
      const float2 qc = *(const float2*)(q + 16);  // y2 pad
      const float dab = dAB[r], dad = dAD[r], ab = pAB[r], ad = pAD[r];
      const float delta = dab*dab*qa.x + dad*dad*qa.y;
      const float gauss = __expf(-0.5f*kGauss*delta);
      const float refined = gauss * qa.z;
      const bool inside = (ab >= kEps) & (ab <= qa.w) &
                          (ad >= kEps) & (ad <= qb.x) & (gauss > kScoreTh);
      const float ltrb = fmaxf(fmaxf(px - qb.y, py - qb.z),
                               fmaxf(qb.w - px, qc.x - py));
      const bool rmask = (ltrb >= rlo) & (ltrb <= rhi);
      const float posm = (inside && rmask) ? qc.y : 0.f;
      const float masked = refined*posm - (1.f - posm);
      if (masked > bestv) { bestv = masked; besti = m; }  // strict > keeps first max
    }
  }

  // merge lane halves: partner lane^16 owns M+8 rows of same anchor column
  {
    const float pv = swapx16f(bestv);
    const int   pi = swapx16i(besti);
    if (pv > bestv || (pv == bestv && pi < besti)) { bestv = pv; besti = pi; }
  }

  // ---------------- per-anchor tail (lanes 0..15 own one anchor) ----------------
  if (lowHalf) {
    const long idx = (long)b*kL + l;
    const bool pos = bestv > 0.f;
    const int  n   = besti;
    const int  bg  = *bg_index_p;
    const int  lbl = pos ? gt_labels[b*kNgt + n] : bg;

    float gb5[5];
    #pragma unroll
    for (int k = 0; k < 5; ++k) gb5[k] = sgRaw[n*5 + k];

    const long BL = (long)Btot * kL;
    out[idx] = (float)lbl;                              // assigned_labels
    float* outBox = out + BL;                           // assigned_boxes
    #pragma unroll
    for (int k = 0; k < 5; ++k)
      outBox[idx*5 + k] = pos ? gb5[k] : 0.f;

    float iou = 0.f;
    if (pos) iou = probiou(pred_boxes + idx*5, gb5);

    float* outSc = out + BL*6;                          // assigned_scores (one-hot * iou)
    #pragma unroll
    for (int c = 0; c < kC; ++c)
      outSc[idx*kC + c] = (pos && (lbl == c)) ? iou : 0.f;
  }
}

extern "C" void kernel_launch(void* const* d_in, const int* in_sizes, int n_in,
                              void* d_out, int out_size, void* d_ws, size_t ws_size,
                              hipStream_t stream) {
  // setup_inputs order:
  // 0 pred_scores (unused), 1 pred_boxes, 2 anchor_points, 3 gt_labels,
  // 4 gt_boxes, 5 pad_gt_mask, 6 bg_index, 7 stride_tensor (unused)
  const float* pred_boxes = (const float*)d_in[1];
  const float* anchors    = (const float*)d_in[2];
  const int*   gt_labels  = (const int*)d_in[3];
  const float* gt_boxes   = (const float*)d_in[4];
  const float* pad_mask   = (const float*)d_in[5];
  const int*   bg_index   = (const int*)d_in[6];
  float* out = (float*)d_out;

  const int B = in_sizes[3] / kNgt;   // gt_labels is (B, 64, 1)
  dim3 grid((unsigned)(B * kBlkPerB)), block(128);
  hipLaunchKernelGGL(rot_gauss_assigner, grid, block, 0, stream,
                     pred_boxes, anchors, gt_labels, gt_boxes, pad_mask,
                     bg_index, out, B);
}